// IGNet_74354473828989
// MI455X (gfx1250) — compile-verified
//
#include <hip/hip_runtime.h>
#include <hip/hip_bf16.h>
#include <math.h>

// ---------------------------------------------------------------------------
// Constants from the reference
// ---------------------------------------------------------------------------
#define NUM_VIEW   300
#define NUM_ANGLE  12
#define NUM_DEPTH  4
#define VA         (NUM_VIEW * NUM_ANGLE)   // 3600
#define GRASP_MAX_WIDTH 0.1f

typedef float v2f __attribute__((ext_vector_type(2)));
typedef float v8f __attribute__((ext_vector_type(8)));

// ---------------------------------------------------------------------------
// K1: generate GRASP_VIEWS (300x3 f32), GRASP_ROT (3600x9 f32), PRED_KP (3600x12 f32)
// One thread per (view, angle). f64 math to match numpy, cast to f32 like reference.
// ---------------------------------------------------------------------------
__global__ void k_const(float* __restrict__ views, float* __restrict__ grot,
                        float* __restrict__ predkp) {
  int idx = blockIdx.x * blockDim.x + threadIdx.x;
  if (idx >= VA) return;
  int v = idx / NUM_ANGLE, a = idx % NUM_ANGLE;

  const double PI  = 3.14159265358979323846;
  const double phi = (sqrt(5.0) - 1.0) * 0.5;
  double z = (2.0 * (double)v + 1.0) / (double)NUM_VIEW - 1.0;
  double r = sqrt(fmax(1.0 - z * z, 0.0));
  double ang = ((2.0 * PI) * (double)v) * phi;
  double x = r * cos(ang), y = r * sin(ang);
  float vx = (float)x, vy = (float)y, vz = (float)z;   // views stored f32
  if (a == 0) { views[v*3+0] = vx; views[v*3+1] = vy; views[v*3+2] = vz; }

  // towards = -views (f32 -> f64), as in reference
  double axx = -(double)vx, axy = -(double)vy, axz = -(double)vz;
  double ayx = -axy, ayy = axx, ayz = 0.0;
  double ny = sqrt(ayx*ayx + ayy*ayy);
  if (ny == 0.0) { ayx = 0.0; ayy = 1.0; ayz = 0.0; ny = 1.0; }
  double nx = sqrt(axx*axx + axy*axy + axz*axz);
  axx /= nx; axy /= nx; axz /= nx;
  ayx /= ny; ayy /= ny; ayz /= ny;
  // axis_z = axis_x x axis_y
  double azx = axy*ayz - axz*ayy;
  double azy = axz*ayx - axx*ayz;
  double azz = axx*ayy - axy*ayx;

  double theta = (PI / (double)NUM_ANGLE) * (double)a;
  double c = cos(theta), s = sin(theta);
  // R = R2 @ R1 ; R2 columns = [ax, ay, az]; R1 = rot about x
  float R[9];
  R[0] = (float)axx; R[1] = (float)(ayx*c + azx*s); R[2] = (float)(azx*c - ayx*s);
  R[3] = (float)axy; R[4] = (float)(ayy*c + azy*s); R[5] = (float)(azy*c - ayy*s);
  R[6] = (float)axz; R[7] = (float)(ayz*c + azz*s); R[8] = (float)(azz*c - ayz*s);
  #pragma unroll
  for (int k = 0; k < 9; k++) grot[(size_t)idx*9 + k] = R[k];

  // PRED_KP: keypoints with widths=depths=0.02 (f32 math on f32 R)
  const float lx[4] = {0.02f, 0.02f, 0.02f, -0.02f};
  const float ly[4] = {0.0f, 0.01f, -0.01f, 0.0f};
  #pragma unroll
  for (int k = 0; k < 4; k++)
    #pragma unroll
    for (int i = 0; i < 3; i++)
      predkp[(size_t)idx*12 + k*3 + i] = R[i*3+0]*lx[k] + R[i*3+1]*ly[k];
}

// ---------------------------------------------------------------------------
// K2: gv_trans[b][j] = R_b @ views[j];  gp_trans[b][p] = R_b @ gp[p] + t_b
// ---------------------------------------------------------------------------
__global__ void k_transform(const float* __restrict__ views,
                            const float* __restrict__ pose,
                            const float* __restrict__ gp,
                            float* __restrict__ gv, float* __restrict__ gpt,
                            int B, int Np) {
  int per = NUM_VIEW + Np;
  int idx = blockIdx.x * blockDim.x + threadIdx.x;
  if (idx >= B * per) return;
  int b = idx / per, r = idx % per;
  const float* P = pose + (size_t)b * 12;
  if (r < NUM_VIEW) {
    const float* vv = views + r*3;
    #pragma unroll
    for (int i = 0; i < 3; i++)
      gv[((size_t)b*NUM_VIEW + r)*3 + i] =
        P[i*4+0]*vv[0] + P[i*4+1]*vv[1] + P[i*4+2]*vv[2];
  } else {
    int p = r - NUM_VIEW;
    const float* g = gp + ((size_t)b*Np + p)*3;
    #pragma unroll
    for (int i = 0; i < 3; i++)
      gpt[((size_t)b*Np + p)*3 + i] =
        P[i*4+0]*g[0] + P[i*4+1]*g[1] + P[i*4+2]*g[2] + P[i*4+3];
  }
}

// ---------------------------------------------------------------------------
// K3: view_inds[b][i] = argmin_j || views[i] - gv_trans[b][j] ||^2 (knn1 formula)
// ---------------------------------------------------------------------------
__global__ void k_view_knn(const float* __restrict__ views,
                           const float* __restrict__ gv,
                           int* __restrict__ view_inds, int B) {
  int idx = blockIdx.x * blockDim.x + threadIdx.x;
  if (idx >= B * NUM_VIEW) return;
  int b = idx / NUM_VIEW, i = idx % NUM_VIEW;
  float qx = views[i*3], qy = views[i*3+1], qz = views[i*3+2];
  float qq = qx*qx + qy*qy + qz*qz;
  const float* rb = gv + (size_t)b * NUM_VIEW * 3;
  float best = INFINITY; int bi = 0;
  for (int j = 0; j < NUM_VIEW; j++) {
    float rx = rb[j*3], ry = rb[j*3+1], rz = rb[j*3+2];
    float d = qq - 2.0f*(qx*rx + qy*ry + qz*rz) + (rx*rx + ry*ry + rz*rz);
    if (d < best) { best = d; bi = j; }
  }
  view_inds[idx] = bi;
}

// ---------------------------------------------------------------------------
// K4: angle alignment. Per (b, view): build temp keypoints for the 12 angles of
// the reindexed rotation (R_pose @ GRASP_ROT[view_inds[v], a']), 1-NN in 12-dim
// against PRED_KP (and mirrored version), producing va_map and rot_final.
// ---------------------------------------------------------------------------
#define VPB 10
__global__ void k_angle(const float* __restrict__ grot,
                        const float* __restrict__ predkp,
                        const float* __restrict__ pose,
                        const int* __restrict__ view_inds,
                        int* __restrict__ va_map,
                        float* __restrict__ rot_final, int B) {
  int b = blockIdx.y;
  int vbase = blockIdx.x * VPB;
  int t = threadIdx.x;
  __shared__ float kpS[VPB][12][12];
  __shared__ float rtS[VPB][12][9];
  int lv = t / 12, a = t % 12;
  bool active = (t < VPB * 12) && (vbase + lv < NUM_VIEW);
  const float* P = pose + (size_t)b * 12;
  if (active) {
    int v = vbase + lv;
    int vi = view_inds[b*NUM_VIEW + v];
    const float* G = grot + (size_t)(vi*NUM_ANGLE + a) * 9;
    float Rt[9];
    #pragma unroll
    for (int i = 0; i < 3; i++)
      #pragma unroll
      for (int j = 0; j < 3; j++)
        Rt[i*3+j] = P[i*4+0]*G[0*3+j] + P[i*4+1]*G[1*3+j] + P[i*4+2]*G[2*3+j];
    #pragma unroll
    for (int k = 0; k < 9; k++) rtS[lv][a][k] = Rt[k];
    const float lx[4] = {0.02f, 0.02f, 0.02f, -0.02f};
    const float ly[4] = {0.0f, 0.01f, -0.01f, 0.0f};
    #pragma unroll
    for (int k = 0; k < 4; k++)
      #pragma unroll
      for (int i = 0; i < 3; i++)
        kpS[lv][a][k*3+i] = Rt[i*3+0]*lx[k] + Rt[i*3+1]*ly[k];
  }
  __syncthreads();
  if (!active) return;
  int v = vbase + lv;
  float q[12]; float qq = 0.0f;
  const float* Q = predkp + (size_t)(v*NUM_ANGLE + a) * 12;
  #pragma unroll
  for (int j = 0; j < 12; j++) { q[j] = Q[j]; qq += q[j]*q[j]; }
  const int sym[12] = {0,1,2, 6,7,8, 3,4,5, 9,10,11};
  float d1 = INFINITY; int i1 = 0;
  float d2 = INFINITY; int i2 = 0;
  for (int ap = 0; ap < 12; ap++) {
    float dot = 0.0f, rr = 0.0f;
    #pragma unroll
    for (int j = 0; j < 12; j++) { float rv = kpS[lv][ap][j]; dot += q[j]*rv; rr += rv*rv; }
    float d = qq - 2.0f*dot + rr;
    if (d < d1) { d1 = d; i1 = ap; }
  }
  for (int ap = 0; ap < 12; ap++) {
    float dot = 0.0f, rr = 0.0f;
    #pragma unroll
    for (int j = 0; j < 12; j++) { float rv = kpS[lv][ap][sym[j]]; dot += q[j]*rv; rr += rv*rv; }
    float d = qq - 2.0f*dot + rr;
    if (d < d2) { d2 = d; i2 = ap; }
  }
  int ind = (d1 < d2) ? i1 : i2;
  int vi = view_inds[b*NUM_VIEW + v];
  va_map[b*VA + v*NUM_ANGLE + a] = vi*NUM_ANGLE + ind;
  float* RF = rot_final + ((size_t)b*VA + v*NUM_ANGLE + a) * 9;
  #pragma unroll
  for (int k = 0; k < 9; k++) RF[k] = rtS[lv][ind][k];
}

// ---------------------------------------------------------------------------
// K5: seed->grasp-point 1-NN via V_WMMA_F32_16X16X4_F32.
// 256-thread block (8 waves). All Np points staged ONCE into LDS as
// float4(x,y,z,|r|^2) (padded to 16 with rr=1e30 sentinels), then each wave
// owns a 16-row query strip and loops 16-col tiles: ds_load_b128 + WMMA.
// A layout (16x4 f32): lane l: M=l%16, VGPR0->K=(l<16?0:2), VGPR1->K+1.
// B layout (4x16 f32): lane l: N=l%16, VGPR0->K=(l<16?0:2), VGPR1->K+1.
// C layout: lane l, vgpr v: M = v + 8*(l>=16), N = l%16.
// Argmin reduced across the 16 lanes of each half with shfl_xor butterflies.
// ---------------------------------------------------------------------------
__global__ void k_seed_knn(const float* __restrict__ seed,
                           const float* __restrict__ gpt,
                           int* __restrict__ nn, int B, int Ns, int Np) {
  extern __shared__ char smem[];
  const int tid  = threadIdx.x;
  const int wave = tid >> 5, lw = tid & 31;
  const int half = lw >> 4, ml = lw & 15;
  const int kb = half * 2;
  const int b = blockIdx.y;
  const int NpPad = (Np + 15) & ~15;

  float4* rpts = (float4*)smem;                          // NpPad float4s
  float*  qqS  = (float*)(smem + (size_t)NpPad * 16);    // 8 waves x 16 floats

  // Stage all grasp points (x, y, z, |r|^2) into LDS; pad with sentinels.
  const float* rp = gpt + (size_t)b * Np * 3;
  for (int j = tid; j < NpPad; j += blockDim.x) {
    float4 e;
    if (j < Np) {
      float x = rp[j*3], y = rp[j*3+1], z = rp[j*3+2];
      e = make_float4(x, y, z, x*x + y*y + z*z);
    } else {
      e = make_float4(0.0f, 0.0f, 0.0f, 1e30f);
    }
    rpts[j] = e;
  }

  // Per-wave query strip: A operand + query norms.
  const int tileBase = blockIdx.x * 128 + wave * 16;
  int qrow = tileBase + ml; if (qrow >= Ns) qrow = Ns - 1;
  const float* q3 = seed + ((size_t)b*Ns + qrow) * 3;
  v2f A; A.x = q3[kb]; A.y = (kb == 0) ? q3[1] : 0.0f;   // K=3 slot padded with 0
  if (lw < 16)
    qqS[wave*16 + lw] = q3[0]*q3[0] + q3[1]*q3[1] + q3[2]*q3[2];
  __syncthreads();

  float myqq[8];
  #pragma unroll
  for (int v = 0; v < 8; v++) myqq[v] = qqS[wave*16 + v + half*8];

  float best[8]; int bj[8];
  #pragma unroll
  for (int v = 0; v < 8; v++) { best[v] = INFINITY; bj[v] = 0; }

  for (int t = 0; t < NpPad; t += 16) {
    float4 r4 = rpts[t + ml];
    v2f Bm; Bm.x = (kb == 0) ? r4.x : r4.z; Bm.y = (kb == 0) ? r4.y : 0.0f;
    v8f c = {};
    c = __builtin_amdgcn_wmma_f32_16x16x4_f32(false, A, false, Bm, (short)0, c,
                                              false, false);
    int j = t + ml;
    #pragma unroll
    for (int v = 0; v < 8; v++) {
      float d = myqq[v] - 2.0f * c[v] + r4.w;
      if (d < best[v]) { best[v] = d; bj[v] = j; }
    }
  }

  // Reduce across the 16 column-slots (lanes of each half): first-min semantics
  // (on ties, smaller j wins).
  #pragma unroll
  for (int m = 1; m <= 8; m <<= 1) {
    #pragma unroll
    for (int v = 0; v < 8; v++) {
      float ov = __shfl_xor(best[v], m, 32);
      int   oj = __shfl_xor(bj[v],  m, 32);
      if (ov < best[v] || (ov == best[v] && oj < bj[v])) { best[v] = ov; bj[v] = oj; }
    }
  }
  if (ml == 0) {
    #pragma unroll
    for (int v = 0; v < 8; v++) {
      int M = v + half * 8;
      int row = tileBase + M;
      if (row < Ns) nn[b*Ns + row] = bj[v];
    }
  }
}

// ---------------------------------------------------------------------------
// K6: per (b, seed): graspness over all 3600 (v,a), normalized per row; plus
// pts, top_rot, top_widths, and staged (masked) top scores.
// ---------------------------------------------------------------------------
__global__ void k_main(const float* __restrict__ labels,
                       const float* __restrict__ offsets,
                       const float* __restrict__ gpt,
                       const float* __restrict__ rot_final,
                       const int* __restrict__ va_map,
                       const int* __restrict__ nn,
                       const int* __restrict__ topinds,
                       float* __restrict__ o_rot, float* __restrict__ o_w,
                       float* __restrict__ o_g, float* __restrict__ o_p,
                       float* __restrict__ top_stage,
                       int B, int Ns, int Np) {
  int b = blockIdx.y, n = blockIdx.x, tid = threadIdx.x;
  __shared__ float g[VA];
  __shared__ float redmin[256], redmax[256];
  __shared__ float gmm[2];

  int p = nn[b*Ns + n];
  const float* lab = labels + ((size_t)b*Np + p) * (size_t)VA * 4;
  const int* vm = va_map + (size_t)b*VA;

  float lmin = INFINITY, lmax = -INFINITY;
  for (int va = tid; va < VA; va += blockDim.x) {
    int gidx = vm[va];
    const float4 s4 = *reinterpret_cast<const float4*>(lab + (size_t)gidx * 4);
    int c = 0;
    c += (s4.x > 0.0f && s4.x <= 0.6f);
    c += (s4.y > 0.0f && s4.y <= 0.6f);
    c += (s4.z > 0.0f && s4.z <= 0.6f);
    c += (s4.w > 0.0f && s4.w <= 0.6f);
    float gv = (float)c * 0.25f;
    g[va] = gv;
    lmin = fminf(lmin, gv); lmax = fmaxf(lmax, gv);
  }
  redmin[tid] = lmin; redmax[tid] = lmax;
  __syncthreads();
  if (tid == 0) {
    float a = INFINITY, m = -INFINITY;
    for (int i = 0; i < (int)blockDim.x; i++) { a = fminf(a, redmin[i]); m = fmaxf(m, redmax[i]); }
    gmm[0] = a; gmm[1] = m;
  }
  __syncthreads();
  float denom = gmm[1] - gmm[0] + 1e-8f;
  float gmin = gmm[0];
  float* gout = o_g + ((size_t)(b*Ns + n)) * VA;
  for (int va = tid; va < VA; va += blockDim.x)
    gout[va] = (g[va] - gmin) / denom;

  int gi = topinds[b*Ns + n];
  if (tid < 9)
    o_rot[(size_t)(b*Ns + n)*9 + tid] = rot_final[((size_t)b*VA + gi)*9 + tid];
  if (tid < 3)
    o_p[(size_t)(b*Ns + n)*3 + tid] = gpt[((size_t)b*Np + p)*3 + tid];
  if (tid < 4) {
    int gidx2 = vm[gi];
    float s = lab[(size_t)gidx2*4 + tid];
    float w = offsets[(((size_t)b*Np + p)*VA + gidx2)*4 + tid];
    o_w[(size_t)(b*Ns + n)*4 + tid] = w;
    top_stage[(size_t)(b*Ns + n)*4 + tid] = (s > 0.0f && w <= GRASP_MAX_WIDTH) ? s : 0.0f;
  }
}

// ---------------------------------------------------------------------------
// K7: global log-normalization of top scores (single block).
// ---------------------------------------------------------------------------
__global__ void k_topnorm(const float* __restrict__ stage,
                          float* __restrict__ o_sc, int N) {
  __shared__ float rmax[256], rmin[256];
  __shared__ float um[2];
  int tid = threadIdx.x;
  float lmax = -INFINITY, lmin = INFINITY;
  for (int i = tid; i < N; i += 256) {
    float s = stage[i];
    lmax = fmaxf(lmax, s);
    if (s > 0.0f) lmin = fminf(lmin, s);
  }
  rmax[tid] = lmax; rmin[tid] = lmin;
  __syncthreads();
  if (tid == 0) {
    float a = -INFINITY, c = INFINITY;
    for (int i = 0; i < 256; i++) { a = fmaxf(a, rmax[i]); c = fminf(c, rmin[i]); }
    um[0] = a; um[1] = c;
  }
  __syncthreads();
  float umax = um[0], umin = um[1];
  float denom = logf(umax / umin) + 1e-8f;
  for (int i = tid; i < N; i += 256) {
    float s = stage[i];
    o_sc[i] = (s > 0.0f) ? (logf(umax / s) / denom) : s;
  }
}

// ---------------------------------------------------------------------------
// Host launcher
// ---------------------------------------------------------------------------
extern "C" void kernel_launch(void* const* d_in, const int* in_sizes, int n_in,
                              void* d_out, int out_size, void* d_ws, size_t ws_size,
                              hipStream_t stream) {
  const float* pc     = (const float*)d_in[0];   // (B,Ns,3)
  const float* pose   = (const float*)d_in[1];   // (B,3,4)
  const float* gp     = (const float*)d_in[2];   // (B,Np,3)
  const float* labels = (const float*)d_in[3];   // (B,Np,300,12,4)
  const float* offs   = (const float*)d_in[4];   // (B,Np,300,12,4)
  const int*   tinds  = (const int*)d_in[5];     // (B,Ns)

  const int B  = in_sizes[1] / 12;
  const int Ns = in_sizes[0] / (3 * B);
  const int Np = in_sizes[2] / (3 * B);

  // workspace carving (256B aligned regions)
  char* w = (char*)d_ws;
  auto take = [&](size_t bytes) -> void* {
    void* p = (void*)w;
    w += (bytes + 255) & ~((size_t)255);
    return p;
  };
  float* ws_views  = (float*)take((size_t)NUM_VIEW * 3 * sizeof(float));
  float* ws_grot   = (float*)take((size_t)VA * 9 * sizeof(float));
  float* ws_predkp = (float*)take((size_t)VA * 12 * sizeof(float));
  float* ws_gv     = (float*)take((size_t)B * NUM_VIEW * 3 * sizeof(float));
  float* ws_gpt    = (float*)take((size_t)B * Np * 3 * sizeof(float));
  float* ws_rotf   = (float*)take((size_t)B * VA * 9 * sizeof(float));
  float* ws_stage  = (float*)take((size_t)B * Ns * 4 * sizeof(float));
  int*   ws_vinds  = (int*)take((size_t)B * NUM_VIEW * sizeof(int));
  int*   ws_vamap  = (int*)take((size_t)B * VA * sizeof(int));
  int*   ws_nn     = (int*)take((size_t)B * Ns * sizeof(int));

  // output carving: top_rot, top_scores, top_widths, graspness, pts
  float* out = (float*)d_out;
  const size_t BN = (size_t)B * Ns;
  float* o_rot = out;
  float* o_sc  = o_rot + BN * 9;
  float* o_w   = o_sc + BN * 4;
  float* o_g   = o_w + BN * 4;
  float* o_p   = o_g + BN * VA;

  // 1: constants
  k_const<<<(VA + 255) / 256, 256, 0, stream>>>(ws_views, ws_grot, ws_predkp);
  // 2: transforms
  {
    int total = B * (NUM_VIEW + Np);
    k_transform<<<(total + 255) / 256, 256, 0, stream>>>(ws_views, pose, gp,
                                                         ws_gv, ws_gpt, B, Np);
  }
  // 3: view knn
  {
    int total = B * NUM_VIEW;
    k_view_knn<<<(total + 127) / 128, 128, 0, stream>>>(ws_views, ws_gv, ws_vinds, B);
  }
  // 4: angle alignment
  k_angle<<<dim3((NUM_VIEW + VPB - 1) / VPB, B), 128, 0, stream>>>(
      ws_grot, ws_predkp, pose, ws_vinds, ws_vamap, ws_rotf, B);
  // 5: seed knn (WMMA, LDS-staged points, 8 waves/block)
  {
    int NpPad = (Np + 15) & ~15;
    size_t shmem = (size_t)NpPad * 16 + 8 * 16 * sizeof(float);
    k_seed_knn<<<dim3((Ns + 127) / 128, B), 256, shmem, stream>>>(pc, ws_gpt,
                                                                  ws_nn, B, Ns, Np);
  }
  // 6: graspness + gathers
  k_main<<<dim3(Ns, B), 256, 0, stream>>>(labels, offs, ws_gpt, ws_rotf, ws_vamap,
                                          ws_nn, tinds, o_rot, o_w, o_g, o_p,
                                          ws_stage, B, Ns, Np);
  // 7: global top-score normalization
  k_topnorm<<<1, 256, 0, stream>>>(ws_stage, o_sc, (int)(BN * 4));
}